// MultiScaleTrendDirectionLoss_34256659153510
// MI455X (gfx1250) — compile-verified
//
#include <hip/hip_runtime.h>
#include <math.h>

// ---------------------------------------------------------------------------
// MultiScaleTrendDirectionLoss on gfx1250 (MI455X, wave32)
//
// EMA scan done as chunked matmul on the WMMA pipe:
//   ema[t0+i] = sum_j W[i][j] * x[t0+j]  +  carry * q^(i+1),
//   W[i][j] = alpha * q^(i-j) (lower triangular 16x16), q = 1-alpha
// -> 4x V_WMMA_F32_16X16X4_F32 per 16x16 tile per EMA stream.
// Segment-parallel over S with a 3-kernel carry-decomposition pipeline.
// ---------------------------------------------------------------------------

typedef float v2f __attribute__((ext_vector_type(2)));
typedef float v8f __attribute__((ext_vector_type(8)));

#define B_ROWS   2048
#define S_LEN    8192
#define NALPHA   3
#define NSEG     32
#define SEGLEN   256           // 16 chunks of 16 timesteps
#define CHUNKS   16
#define NGROUP   (B_ROWS / 16) // 128 row-groups of 16 rows
#define WPB      8             // waves per block (256 threads)

__device__ __constant__ float g_alpha[NALPHA]  = {0.5f, 0.2f, 0.05f};
__device__ __constant__ float g_weight[NALPHA] = {0.5f, 0.3f, 0.2f};

__device__ __forceinline__ float safef(float v) {
    v = __builtin_isnan(v) ? 0.0f : v;            // nan -> 0
    return fminf(fmaxf(v, -1.0e6f), 1.0e6f);      // +/-inf clamp to +/-1e6
}

// PHASE 1: local scans with zero carry, write segment-end values to ws_l.
// PHASE 3: scans with true carries (ws_c), Huber loss, atomicAdd into out.
template <int PHASE>
__global__ __launch_bounds__(256) void ema_scan_kernel(
    const float* __restrict__ pred, const float* __restrict__ target,
    float* __restrict__ ws_l, const float* __restrict__ ws_c,
    float* __restrict__ out)
{
    const int lane  = threadIdx.x & 31;  // wave32
    const int wid   = threadIdx.x >> 5;
    const int unit  = blockIdx.x * WPB + wid;     // 0..NGROUP*NSEG-1
    const int group = unit / NSEG;
    const int seg   = unit % NSEG;
    const int row0  = group * 16;
    const int t0    = seg * SEGLEN;

    const int mrow = lane & 15;   // N (row) index / A-matrix M index
    const int hi   = lane >> 4;   // lane half selects K sub-pair

    __shared__ float tiles[WPB][2][16][17];       // +1 pad vs bank conflicts

    // --- constant A fragments: W_alpha[i][j] = alpha*q^(i-j), lower-tri ---
    // A 16x4 layout: lanes 0-15 M=lane K={4s,4s+1}; lanes 16-31 K={4s+2,4s+3}
    v2f   afrag[NALPHA][4];
    float qpw[NALPHA][8];         // q^(d+1) (lo half) / q^(d+9) (hi half)
    for (int a = 0; a < NALPHA; ++a) {
        const float al = g_alpha[a], q = 1.0f - al;
        for (int s = 0; s < 4; ++s) {
            const int k0 = 4 * s + 2 * hi;
            float w0 = 0.0f, w1 = 0.0f;
            if (k0 <= mrow)     { w0 = al; for (int e = 0; e < mrow - k0;     ++e) w0 *= q; }
            if (k0 + 1 <= mrow) { w1 = al; for (int e = 0; e < mrow - k0 - 1; ++e) w1 *= q; }
            v2f f; f.x = w0; f.y = w1;
            afrag[a][s] = f;
        }
        float base = q;
        if (hi) for (int e = 0; e < 8; ++e) base *= q;   // q^9 for upper half
        for (int d = 0; d < 8; ++d) { qpw[a][d] = base; base *= q; }
    }

    // --- carries per (alpha, input); lane-resident for row = lane&15 ---
    float cp[NALPHA], ct[NALPHA];
    for (int a = 0; a < NALPHA; ++a) {
        if (PHASE == 1) { cp[a] = 0.0f; ct[a] = 0.0f; }
        else {
            cp[a] = ws_c[((size_t)(0 * NALPHA + a) * NSEG + seg) * B_ROWS + row0 + mrow];
            ct[a] = ws_c[((size_t)(1 * NALPHA + a) * NSEG + seg) * B_ROWS + row0 + mrow];
        }
    }
    float acc[NALPHA] = {0.0f, 0.0f, 0.0f};

    for (int ck = 0; ck < CHUNKS; ++ck) {
        const int tc = t0 + ck * 16;

        // coalesced tile load: 16 rows x 16 cols, 2x float4 per lane
        for (int rep = 0; rep < 2; ++rep) {
            const int id = lane + rep * 32;       // 0..63
            const int r = id >> 2, cg = id & 3;
            const size_t base = (size_t)(row0 + r) * S_LEN + tc + cg * 4;
            const float4 vp = *reinterpret_cast<const float4*>(pred + base);
            const float4 vt = *reinterpret_cast<const float4*>(target + base);
            float* dp = &tiles[wid][0][r][cg * 4];
            float* dt = &tiles[wid][1][r][cg * 4];
            dp[0] = safef(vp.x); dp[1] = safef(vp.y); dp[2] = safef(vp.z); dp[3] = safef(vp.w);
            dt[0] = safef(vt.x); dt[1] = safef(vt.y); dt[2] = safef(vt.z); dt[3] = safef(vt.w);
            if (ck + 1 < CHUNKS) {                // emits global_prefetch_b8
                __builtin_prefetch(pred + base + 16, 0, 3);
                __builtin_prefetch(target + base + 16, 0, 3);
            }
        }
        __syncthreads();

        // B fragments: B[k][n] = x[row n][tc+k]; lanes hold n=lane&15,
        // K sub-pair {4s+2hi, 4s+2hi+1}
        v2f bp[4], bt[4];
        #pragma unroll
        for (int s = 0; s < 4; ++s) {
            const int k = 4 * s + 2 * hi;
            v2f p; p.x = tiles[wid][0][mrow][k]; p.y = tiles[wid][0][mrow][k + 1];
            v2f t; t.x = tiles[wid][1][mrow][k]; t.y = tiles[wid][1][mrow][k + 1];
            bp[s] = p; bt[s] = t;
        }
        __syncthreads();

        for (int a = 0; a < NALPHA; ++a) {
            v8f dp = {0.f,0.f,0.f,0.f,0.f,0.f,0.f,0.f};
            v8f dt = {0.f,0.f,0.f,0.f,0.f,0.f,0.f,0.f};
            #pragma unroll
            for (int s = 0; s < 4; ++s) {
                dp = __builtin_amdgcn_wmma_f32_16x16x4_f32(
                        false, afrag[a][s], false, bp[s], (short)0, dp, false, false);
                dt = __builtin_amdgcn_wmma_f32_16x16x4_f32(
                        false, afrag[a][s], false, bt[s], (short)0, dt, false, false);
            }
            // E[d] = D[d] + carry * q^(M+1); VGPR d holds M=d (lo) / M=d+8 (hi)
            float Ep[8], Et[8];
            #pragma unroll
            for (int d = 0; d < 8; ++d) {
                Ep[d] = dp[d] + cp[a] * qpw[a][d];
                Et[d] = dt[d] + ct[a] * qpw[a][d];
            }
            if (PHASE == 3) {
                // prev for M=0 (lo half) is the carry; for M=8 (hi half) it's
                // E[7] from the matching lo-half lane.
                const float s7p = __shfl(Ep[7], mrow);
                const float s7t = __shfl(Et[7], mrow);
                float prevp = (lane < 16) ? cp[a] : s7p;
                float prevt = (lane < 16) ? ct[a] : s7t;
                #pragma unroll
                for (int d = 0; d < 8; ++d) {
                    const float pd = Ep[d] - prevp;
                    const float td = Et[d] - prevt;
                    prevp = Ep[d]; prevt = Et[d];
                    const float dir = pd * td;
                    const float m   = fmaxf(1.0f - dir, 0.0f);
                    acc[a] += (dir < 0.0f) ? fabsf(pd - td) : 0.5f * m * m;
                }
            }
            // new carry = E[M=15] = E[7] of upper half, per row
            cp[a] = __shfl(Ep[7], 16 + mrow);
            ct[a] = __shfl(Et[7], 16 + mrow);
        }
    }

    if (PHASE == 1) {
        if (lane < 16) {
            for (int a = 0; a < NALPHA; ++a) {
                ws_l[((size_t)(0 * NALPHA + a) * NSEG + seg) * B_ROWS + row0 + lane] = cp[a];
                ws_l[((size_t)(1 * NALPHA + a) * NSEG + seg) * B_ROWS + row0 + lane] = ct[a];
            }
        }
    } else {
        float part = g_weight[0] * acc[0] + g_weight[1] * acc[1] + g_weight[2] * acc[2];
        // remove artificial t=0 term (pd=td=0 -> quad=0.5) once per row per alpha
        if (seg == 0 && lane < 16)
            part -= 0.5f * (g_weight[0] + g_weight[1] + g_weight[2]);
        for (int off = 16; off; off >>= 1) part += __shfl_xor(part, off);
        if (lane == 0)
            atomicAdd(out, part * (1.0f / ((float)(S_LEN - 1) * (float)B_ROWS)));
    }
}

// Inter-segment carry scan: C_0 = x[row,0];  C_s = q^SEGLEN * C_{s-1} + l_{s-1}
__global__ __launch_bounds__(256) void carry_scan_kernel(
    const float* __restrict__ pred, const float* __restrict__ target,
    const float* __restrict__ ws_l, float* __restrict__ ws_c,
    float* __restrict__ out)
{
    const int tid = blockIdx.x * blockDim.x + threadIdx.x;
    if (tid == 0) *out = 0.0f;                      // re-arm accumulator
    if (tid >= 2 * NALPHA * B_ROWS) return;
    const int row = tid % B_ROWS;
    const int a   = (tid / B_ROWS) % NALPHA;
    const int in  = tid / (B_ROWS * NALPHA);
    const float* x = in ? target : pred;
    const float q  = 1.0f - g_alpha[a];
    const float qL = powf(q, (float)SEGLEN);
    float c = safef(x[(size_t)row * S_LEN]);        // ema[-1] := x[0] trick
    for (int s = 0; s < NSEG; ++s) {
        const size_t idx = ((size_t)(in * NALPHA + a) * NSEG + s) * B_ROWS + row;
        ws_c[idx] = c;
        c = qL * c + ws_l[idx];
    }
}

extern "C" void kernel_launch(void* const* d_in, const int* in_sizes, int n_in,
                              void* d_out, int out_size, void* d_ws, size_t ws_size,
                              hipStream_t stream) {
    (void)in_sizes; (void)n_in; (void)out_size; (void)ws_size;
    const float* pred   = (const float*)d_in[0];
    const float* target = (const float*)d_in[1];
    float* out  = (float*)d_out;
    float* ws_l = (float*)d_ws;                                // 1.5 MB
    float* ws_c = ws_l + (size_t)2 * NALPHA * NSEG * B_ROWS;   // 1.5 MB

    const int blocks = (NGROUP * NSEG) / WPB;   // 512 blocks x 256 threads
    ema_scan_kernel<1><<<blocks, 256, 0, stream>>>(pred, target, ws_l, ws_c, out);
    carry_scan_kernel<<<(2 * NALPHA * B_ROWS + 255) / 256, 256, 0, stream>>>(
        pred, target, ws_l, ws_c, out);
    ema_scan_kernel<3><<<blocks, 256, 0, stream>>>(pred, target, ws_l, ws_c, out);
}